// BktModel_39728447488368
// MI455X (gfx1250) — compile-verified
//
#include <hip/hip_runtime.h>
#include <cmath>

#define B_DIM 64
#define T_DIM 1024
#define NF    512
#define NH    512
#define NKC   500
#define M_TOT (B_DIM * T_DIM)

typedef __attribute__((ext_vector_type(2))) float v2f;
typedef __attribute__((ext_vector_type(8))) float v8f;
typedef __attribute__((ext_vector_type(4))) unsigned int v4u;
typedef __attribute__((ext_vector_type(8))) int v8i;
typedef __attribute__((ext_vector_type(4))) int v4i;

#define BLOCK_M 64
#define LDA     516            // A row stride in LDS floats: 512 + 2-dword pad per 256 dwords
#define NCHUNK  16             // W1 columns per chunk
#define AS_FLOATS (BLOCK_M * LDA)
#define BS_FLOATS (NF * NCHUNK)
#define GEMM_LDS_BYTES ((AS_FLOATS + 2 * BS_FLOATS) * sizeof(float))

#if defined(__has_builtin)
#if __has_builtin(__builtin_amdgcn_tensor_load_to_lds) && __has_builtin(__builtin_amdgcn_s_wait_tensorcnt)
#define USE_TDM 1
#endif
#endif
#ifndef USE_TDM
#define USE_TDM 0
#endif

#if USE_TDM
// One TDM 2D tile load: tile_w x tile_h elements (4B), row stride `stride`
// elements, optional LDS padding (pad_amt_code+1 dwords every
// 2^(pad_int_code+1) dwords). Descriptor per CDNA5 ISA 8.3/8.4; 6-arg builtin.
__device__ __forceinline__ void tdm_load_2d(unsigned int lds_off, const void* gptr,
                                            unsigned int tile_w, unsigned int tile_h,
                                            unsigned int stride, int pad_en,
                                            unsigned int pad_int_code,
                                            unsigned int pad_amt_code)
{
  unsigned long long ga = (unsigned long long)(uintptr_t)gptr;
  v4u g0;
  g0.x = 1u;                                              // count=1, user mode
  g0.y = lds_off;                                         // LDS byte address
  g0.z = (unsigned int)ga;                                // global_addr[31:0]
  g0.w = ((unsigned int)(ga >> 32) & 0x01FFFFFFu)         // global_addr[56:32]
         | 0x80000000u;                                   // type=2 ("image")
  v8i g1;
  g1[0] = (int)((2u << 16)                                // data_size=4B
                | (pad_en ? (1u << 20) : 0u)              // pad_enable
                | (pad_int_code << 22)                    // pad_interval
                | (pad_amt_code << 25));                  // pad_amount
  g1[1] = (int)((tile_w & 0xFFFFu) << 16);                // tensor_dim0[15:0]
  g1[2] = (int)(((tile_w >> 16) & 0xFFFFu)                // tensor_dim0[31:16]
                | ((tile_h & 0xFFFFu) << 16));            // tensor_dim1[15:0]
  g1[3] = (int)(((tile_h >> 16) & 0xFFFFu)                // tensor_dim1[31:16]
                | ((tile_w & 0xFFFFu) << 16));            // tile_dim0
  g1[4] = (int)(tile_h & 0xFFFFu);                        // tile_dim1 (tile_dim2=0)
  g1[5] = (int)stride;                                    // tensor_dim0_stride[31:0]
  g1[6] = 0;
  g1[7] = 0;
  v4i z4 = {0, 0, 0, 0};
  v8i z8 = {0, 0, 0, 0, 0, 0, 0, 0};
  __builtin_amdgcn_tensor_load_to_lds(g0, g1, z4, z4, z8, 0);
}
__device__ __forceinline__ unsigned int lds_off32(const void* p) {
  return (unsigned int)(uintptr_t)p;                      // low 32 bits = LDS offset
}
#endif

// ---------------------------------------------------------------------------
// Kernel 1: fused  o = tanh(FM @ W1 + b1) @ W2 + b2   -> o_buf [M_TOT, 2]
// f32 WMMA 16x16x4; A tile + W1 chunks DMA'd into the 320KB WGP LDS by the
// Tensor Data Mover; W1 double-buffered, prefetch overlapped with compute.
// N-chunk loop unrolled x2 so each buffer base is compile-time -> all B
// fragment loads become ds_load with immediate offsets.
// ---------------------------------------------------------------------------
__global__ __launch_bounds__(128) void mlp_head_kernel(
    const float* __restrict__ FM, const float* __restrict__ W1,
    const float* __restrict__ b1, const float* __restrict__ W2,
    const float* __restrict__ b2, float* __restrict__ o_buf)
{
  extern __shared__ float smem[];
  float* As  = smem;                      // [64][516] (2-dword pad each 256 dwords)
  float* Bs0 = smem + AS_FLOATS;          // buffer 0: [NF][NCHUNK]
  float* Bs1 = Bs0 + BS_FLOATS;           // buffer 1

  const int tid  = threadIdx.x;
  const int wave = tid >> 5;
  const int lane = tid & 31;
  const int col  = lane & 15;             // N (or M for A-frag) index within 16
  const int hi   = lane >> 4;             // lane-half selects K pair {0,2}
  const int koff = hi * 2;
  const int m0   = blockIdx.x * BLOCK_M;
  const int wrow = wave * 16;             // 4 waves x 16 rows

  float po0[8], po1[8];
  #pragma unroll
  for (int v = 0; v < 8; ++v) { po0[v] = 0.f; po1[v] = 0.f; }

  const float* arow = As + (wrow + col) * LDA;

#if USE_TDM
  if (wave == 0) {
    // A tile: 512x64 elems, row stride 512, pad 2 dwords per 256 (codes 7,1)
    tdm_load_2d(lds_off32(As), FM + (size_t)m0 * NF, NF, BLOCK_M, NF, 1, 7u, 1u);
    // First W1 chunk into buffer 0
    tdm_load_2d(lds_off32(Bs0), W1, NCHUNK, NF, NH, 0, 0u, 0u);
    __builtin_amdgcn_s_wait_tensorcnt(0);
  }
  __syncthreads();

  // One N-chunk: prefetch next chunk via TDM, 128 WMMAs, fused W2 epilogue.
  auto do_chunk = [&](int nc, const float* Bcur, float* Bnext, bool prefetch) {
    if (wave == 0 && prefetch)
      tdm_load_2d(lds_off32(Bnext), W1 + nc + NCHUNK, NCHUNK, NF, NH, 0, 0u, 0u);

    v8f acc = {0.f, 0.f, 0.f, 0.f, 0.f, 0.f, 0.f, 0.f};
    #pragma unroll
    for (int kk = 0; kk < NF; kk += 4) {
      int pad = (kk >> 8) << 1;                    // compile-time constant
      // A 16x4 frag: VGPR0 = K=kk+2*hi, VGPR1 = K=kk+2*hi+1 (ISA layout)
      v2f a = *(const v2f*)(arow + kk + koff + pad);
      // B 4x16 frag: row striped across lanes within a VGPR, same K pairing
      v2f b;
      b.x = Bcur[(kk + koff)     * NCHUNK + col];
      b.y = Bcur[(kk + koff + 1) * NCHUNK + col];
      acc = __builtin_amdgcn_wmma_f32_16x16x4_f32(
          false, a, false, b, (short)0, acc, false, false);
    }

    // Epilogue: acc[v] holds row (wrow + v + 8*hi), col (nc + col)
    float w2c0 = W2[(nc + col) * 2 + 0];
    float w2c1 = W2[(nc + col) * 2 + 1];
    float bias = b1[nc + col];
    #pragma unroll
    for (int v = 0; v < 8; ++v) {
      float h = tanhf(acc[v] + bias);
      po0[v] += h * w2c0;
      po1[v] += h * w2c1;
    }

    if (wave == 0) __builtin_amdgcn_s_wait_tensorcnt(0);
    __syncthreads();                               // next chunk visible to all
  };

  for (int nc = 0; nc < NH; nc += 2 * NCHUNK) {
    do_chunk(nc,          Bs0, Bs1, true);                     // nc+16 < 512 always
    do_chunk(nc + NCHUNK, Bs1, Bs0, (nc + 2 * NCHUNK) < NH);
  }
#else
  {
    const float4* src = (const float4*)(FM + (size_t)m0 * NF);
    #pragma unroll 4
    for (int j = 0; j < (BLOCK_M * NF / 4) / 128; ++j) {
      int fid = j * 128 + tid;
      int row = fid >> 7;
      int kq  = fid & 127;
      int k4  = kq * 4;
      float4 v = src[row * 128 + kq];
      *(float4*)(As + row * LDA + k4 + ((k4 >= 256) ? 2 : 0)) = v;
    }
  }
  for (int nc = 0; nc < NH; nc += NCHUNK) {
    __syncthreads();
    #pragma unroll 4
    for (int j = 0; j < (NF * NCHUNK / 4) / 128; ++j) {
      int fid = j * 128 + tid;
      int k   = fid >> 2;
      int c4  = fid & 3;
      float4 v = *(const float4*)(W1 + (size_t)k * NH + nc + c4 * 4);
      *(float4*)(Bs0 + k * NCHUNK + c4 * 4) = v;
    }
    __syncthreads();

    v8f acc = {0.f, 0.f, 0.f, 0.f, 0.f, 0.f, 0.f, 0.f};
    for (int kk = 0; kk < NF; kk += 4) {
      int pad = (kk >> 8) << 1;
      v2f a = *(const v2f*)(arow + kk + koff + pad);
      v2f b;
      b.x = Bs0[(kk + koff)     * NCHUNK + col];
      b.y = Bs0[(kk + koff + 1) * NCHUNK + col];
      acc = __builtin_amdgcn_wmma_f32_16x16x4_f32(
          false, a, false, b, (short)0, acc, false, false);
    }
    float w2c0 = W2[(nc + col) * 2 + 0];
    float w2c1 = W2[(nc + col) * 2 + 1];
    float bias = b1[nc + col];
    #pragma unroll
    for (int v = 0; v < 8; ++v) {
      float h = tanhf(acc[v] + bias);
      po0[v] += h * w2c0;
      po1[v] += h * w2c1;
    }
  }
#endif

  // Reduce the 16 column-partials; width-16 keeps the two row sets separate
  #pragma unroll
  for (int v = 0; v < 8; ++v) {
    #pragma unroll
    for (int off = 8; off >= 1; off >>= 1) {
      po0[v] += __shfl_xor(po0[v], off, 16);
      po1[v] += __shfl_xor(po1[v], off, 16);
    }
  }
  if (col == 0) {
    float bb0 = b2[0], bb1 = b2[1];
    #pragma unroll
    for (int v = 0; v < 8; ++v) {
      int row = m0 + wrow + v + 8 * hi;
      o_buf[row * 2 + 0] = po0[v] + bb0;
      o_buf[row * 2 + 1] = po1[v] + bb1;
    }
  }
}

// ---------------------------------------------------------------------------
// Kernel 2: per-batch stable counting sort of timesteps by chain id.
// ---------------------------------------------------------------------------
__global__ void bucket_kernel(const int* __restrict__ kc,
                              int* __restrict__ order, int* __restrict__ offs)
{
  __shared__ int cnt[NKC];
  __shared__ int start[NKC];
  const int b = blockIdx.x;
  for (int k = threadIdx.x; k < NKC; k += blockDim.x) cnt[k] = 0;
  __syncthreads();
  for (int t = threadIdx.x; t < T_DIM; t += blockDim.x)
    atomicAdd(&cnt[kc[b * T_DIM + t]], 1);
  __syncthreads();
  if (threadIdx.x == 0) {
    int s = 0;
    for (int k = 0; k < NKC; ++k) {
      start[k] = s;
      offs[b * (NKC + 1) + k] = s;
      s += cnt[k];
    }
    offs[b * (NKC + 1) + NKC] = s;
    for (int t = 0; t < T_DIM; ++t) {          // stable: time order preserved
      int k = kc[b * T_DIM + t];
      order[b * T_DIM + start[k]++] = t;
    }
  }
}

// ---------------------------------------------------------------------------
// Kernel 3: 32,000 independent per-(batch, chain) HMM recurrences (NS=2).
// ---------------------------------------------------------------------------
__device__ __forceinline__ float lse2(float a, float b) {
  float m = fmaxf(a, b);
  float d = fminf(a, b) - m;
  return m + log1pf(__expf(d));
}

__global__ void hmm_kernel(const int* __restrict__ corr,
                           const int* __restrict__ order,
                           const int* __restrict__ offs,
                           const float* __restrict__ o_buf,
                           const float* __restrict__ trans,
                           const float* __restrict__ obsl,
                           const float* __restrict__ initl,
                           float* __restrict__ out)
{
  int gid = blockIdx.x * blockDim.x + threadIdx.x;
  if (gid >= B_DIM * NKC) return;
  const int b = gid / NKC;
  const int k = gid % NKC;

  float i0 = initl[k * 2 + 0], i1 = initl[k * 2 + 1];
  float li = lse2(i0, i1);
  float la0 = i0 - li, la1 = i1 - li;

  // log_t[i][j] = trans[k,i,j] - logsumexp_i trans[k,i,j]  (normalize axis 1)
  float t00 = trans[k * 4 + 0], t01 = trans[k * 4 + 1];
  float t10 = trans[k * 4 + 2], t11 = trans[k * 4 + 3];
  float l0 = lse2(t00, t10), l1 = lse2(t01, t11);
  float lt00 = t00 - l0, lt01 = t01 - l1, lt10 = t10 - l0, lt11 = t11 - l1;

  float ob00 = obsl[k * 4 + 0], ob01 = obsl[k * 4 + 1];
  float ob10 = obsl[k * 4 + 2], ob11 = obsl[k * 4 + 3];

  const int s0 = offs[b * (NKC + 1) + k];
  const int s1 = offs[b * (NKC + 1) + k + 1];
  for (int p = s0; p < s1; ++p) {
    const int t = order[b * T_DIM + p];
    const int bt = b * T_DIM + t;
    float o0 = o_buf[bt * 2 + 0], o1 = o_buf[bt * 2 + 1];

    float z00 = ob00 + o0, z01 = ob01 - o0;
    float z10 = ob10 + o1, z11 = ob11 - o1;
    float ls0 = lse2(z00, z01), ls1 = lse2(z10, z11);
    float lo00 = z00 - ls0, lo01 = z01 - ls0;
    float lo10 = z10 - ls1, lo11 = z11 - ls1;

    float p0 = lse2(lo00 + la0, lo10 + la1);
    float p1 = lse2(lo01 + la0, lo11 + la1);
    float pn = lse2(p0, p1);
    out[bt * 2 + 0] = p0 - pn;
    out[bt * 2 + 1] = p1 - pn;

    const int y = corr[bt];
    float lp0 = y ? lo01 : lo00;
    float lp1 = y ? lo11 : lo10;
    float na0 = lse2(lp0 + la0 + lt00, lp1 + la1 + lt01);
    float na1 = lse2(lp0 + la0 + lt10, lp1 + la1 + lt11);
    la0 = na0; la1 = na1;
  }
}

// ---------------------------------------------------------------------------
extern "C" void kernel_launch(void* const* d_in, const int* in_sizes, int n_in,
                              void* d_out, int out_size, void* d_ws, size_t ws_size,
                              hipStream_t stream) {
  const int*   corr  = (const int*)d_in[0];
  const int*   kc    = (const int*)d_in[1];
  const float* FM    = (const float*)d_in[2];
  const float* W1    = (const float*)d_in[3];
  const float* b1    = (const float*)d_in[4];
  const float* W2    = (const float*)d_in[5];
  const float* b2    = (const float*)d_in[6];
  const float* trans = (const float*)d_in[7];
  const float* obsl  = (const float*)d_in[8];
  const float* initl = (const float*)d_in[9];
  float* out = (float*)d_out;

  float* o_buf = (float*)d_ws;                        // M_TOT*2 floats (512 KB)
  int*   order = (int*)(o_buf + (size_t)M_TOT * 2);   // M_TOT ints    (256 KB)
  int*   offs  = order + M_TOT;                       // B*(NKC+1) ints

  mlp_head_kernel<<<M_TOT / BLOCK_M, 128, GEMM_LDS_BYTES, stream>>>(
      FM, W1, b1, W2, b2, o_buf);
  bucket_kernel<<<B_DIM, 256, 0, stream>>>(kc, order, offs);
  hmm_kernel<<<(B_DIM * NKC + 255) / 256, 256, 0, stream>>>(
      corr, order, offs, o_buf, trans, obsl, initl, out);
}